// LensedGalaxyTileDecoder_59708635348992
// MI455X (gfx1250) — compile-verified
//
#include <hip/hip_runtime.h>
#include <math.h>

#define N_TILES 4096
#define SLEN    53
#define NPIX    (SLEN * SLEN)   // 2809
#define TPB     256             // 8 wave32s per block / tile

#if __has_builtin(__builtin_amdgcn_global_load_async_to_lds_b32) && \
    __has_builtin(__builtin_amdgcn_s_wait_asynccnt)
#define HAVE_ASYNC_LDS 1
#else
#define HAVE_ASYNC_LDS 0
#endif

typedef __attribute__((address_space(1))) int* gptr_i32;
typedef __attribute__((address_space(3))) int* lptr_i32;

__global__ __launch_bounds__(TPB) void lensed_tile_decoder_kernel(
    const float* __restrict__ lens_params,   // [4096, 12] : gal0..gal6, b, cx, cy, e1, e2
    const float* __restrict__ gal_bools,     // [4096]
    float* __restrict__ out)                 // [4096, 53, 53]
{
    const int tile = blockIdx.x;
    const float* __restrict__ p = lens_params + tile * 12;

    // gfx1250: global_prefetch_b8 of the per-tile parameter line
    __builtin_prefetch(p, 0, 3);

    // ---- per-tile scalars (uniform addresses -> SMEM scalar loads) ----
    const float flux  = p[0] * 1000.0f + 100.0f;
    const float sigma = p[3] * 3.0f + 1.0f;
    const float s2    = sigma * sigma;
    const float amp   = flux / (2.0f * 3.14159265358979323846f * s2);

    const float b  = p[7];
    const float cx = p[8];
    const float cy = p[9];
    const float e1 = p[10];
    const float e2 = p[11];

    const float ell    = sqrtf(e1 * e1 + e2 * e2);
    const float q      = (1.0f - ell) / (1.0f + ell);
    const float invq   = 1.0f / q;
    const float phirad = atanf(e2 / e1);
    float sinp, cosp;
    sincosf(phirad, &sinp, &cosp);
    const float qfact       = sqrtf(invq - q);       // sqrt(1/q - q)
    const bool  tangential  = (qfact >= 0.001f);     // uniform per tile
    const float safe        = fmaxf(qfact, 0.001f);
    const float b_over_safe = b / safe;
    const float boolv       = gal_bools[tile];

    // ---- per-tile LDS tables: separable Gaussian + lens grid coords ----
    __shared__ float t_tab[SLEN];   // exp(-0.5*(k-26)^2 / s2)
    __shared__ float c_tab[SLEN];   // 53*k/52 - 27
    for (int k = (int)threadIdx.x; k < SLEN; k += TPB) {
        const float d = (float)k - 26.0f;
        t_tab[k] = expf((-0.5f * (d * d)) / s2);
        c_tab[k] = (53.0f * (float)k) / 52.0f - 27.0f;
    }

#if HAVE_ASYNC_LDS
    // gfx1250 async path: stage the 12 params into LDS via
    // GLOBAL_LOAD_ASYNC_TO_LDS_B32 + s_wait_asynccnt (ASYNCcnt).
    __shared__ float stage[16];
    if (threadIdx.x < 12) {
        __builtin_amdgcn_global_load_async_to_lds_b32(
            (gptr_i32)(p + threadIdx.x),
            (lptr_i32)(stage + threadIdx.x),
            0, 0);
    }
    __builtin_amdgcn_s_wait_asynccnt(0);
#endif

    __syncthreads();

    // ---- pixel loop: 2809 pixels, 11 strided iterations ----
    float* __restrict__ o = out + (size_t)tile * NPIX;
    for (int idx = (int)threadIdx.x; idx < NPIX; idx += TPB) {
        const int i = idx / SLEN;            // row  -> y
        const int j = idx - i * SLEN;        // col  -> x
        const float x = c_tab[j];
        const float y = c_tab[i];

        // SIE deflection
        const float dx    = x - cx;
        const float dy    = y - cy;
        const float xsie  = dx * cosp + dy * sinp;
        const float ysie  = dy * cosp - dx * sinp;
        const float r_ell = sqrtf(q * xsie * xsie + ysie * ysie * invq + 1e-12f);
        const float denom = (r_ell == 0.0f) ? 1.0f : r_ell;   // r_ell + (r_ell==0)
        const float rden  = 1.0f / denom;
        const float xr    = xsie * rden;
        const float yr    = ysie * rden;

        float xtg, ytg;
        if (tangential) {                     // uniform branch per tile
            xtg = b_over_safe * atanf(safe * xr);
            ytg = b_over_safe * atanhf(safe * yr);   // |arg| < 1 guaranteed
        } else {
            xtg = b * xr;
            ytg = b * yr;
        }
        const float xg = xtg * cosp - ytg * sinp;
        const float yg = ytg * cosp + xtg * sinp;

        // sample coords (pixel units)
        const float xs = x - xg + 26.0f;
        const float ys = y - yg + 26.0f;

        // clamped bilinear of separable image: factorized form
        const float fx = floorf(xs);
        const float fy = floorf(ys);
        const int x0 = min(max((int)fx,     0), SLEN - 1);
        const int x1 = min(max((int)fx + 1, 0), SLEN - 1);
        const int y0 = min(max((int)fy,     0), SLEN - 1);
        const int y1 = min(max((int)fy + 1, 0), SLEN - 1);
        const float x0f = (float)x0, x1f = (float)x1;
        const float y0f = (float)y0, y1f = (float)y1;

        const float gx = (x1f - xs) * t_tab[x0] + (xs - x0f) * t_tab[x1];
        const float gy = (y1f - ys) * t_tab[y0] + (ys - y0f) * t_tab[y1];

        o[idx] = amp * gx * gy * boolv;
    }
}

extern "C" void kernel_launch(void* const* d_in, const int* in_sizes, int n_in,
                              void* d_out, int out_size, void* d_ws, size_t ws_size,
                              hipStream_t stream) {
    const float* lens_params = (const float*)d_in[0];   // 4096*12 fp32
    const float* gal_bools   = (const float*)d_in[1];   // 4096 fp32
    float* out = (float*)d_out;                          // 4096*2809 fp32

    lensed_tile_decoder_kernel<<<N_TILES, TPB, 0, stream>>>(lens_params, gal_bools, out);
}